// HierarchicalReconstructionLoss_33672543601247
// MI455X (gfx1250) — compile-verified
//
#include <hip/hip_runtime.h>

typedef __attribute__((ext_vector_type(2))) float v2f;
typedef __attribute__((ext_vector_type(8))) float v8f;

// ---------------------------------------------------------------------------
// Zero the 8 per-level accumulators (Scc, Sl2 per level) in workspace.
// ---------------------------------------------------------------------------
__global__ void hrl_init_acc(float* acc) {
    if (threadIdx.x < 8) acc[threadIdx.x] = 0.f;
}

// ---------------------------------------------------------------------------
// Fused per-level kernel: one wave computes one 16x16 output tile.
//   - loads halo'd I (=y_hat_l) and J (=downsampled y) tiles into LDS
//   - horizontal win-sums in VALU directly into WMMA B-operand lane layout
//   - vertical win-sums via chained V_WMMA_F32_16X16X4_F32 with a banded
//     ones matrix S_v built per-lane (A-operand layout: M=lane%16,
//     K=2*(lane/16)+vgpr within each 4-wide chunk)
//   - per-pixel cc + squared diff, wave32 shuffle reduction, atomicAdd
// ---------------------------------------------------------------------------
template<int WIN, int SCALE>
__global__ __launch_bounds__(256)
void hrl_level_kernel(const float* __restrict__ I,   // y_hat_l  [B,H,H]
                      const float* __restrict__ Y,   // y        [B,YH,YH]
                      int H, int tpr, int total_tiles,
                      float* __restrict__ accCC,
                      float* __restrict__ accL2) {
    constexpr int   P   = WIN / 2;          // halo
    constexpr int   HD  = 16 + 2 * P;       // halo'd tile dim (<= 24)
    constexpr int   CH  = (HD + 3) / 4;     // K-chunks of 4 for WMMA
    constexpr float WS  = (float)(WIN * WIN);
    constexpr int   YH  = 512;              // source resolution of y

    __shared__ float sI[8][HD * HD];
    __shared__ float sJ[8][HD * HD];

    const int lane = threadIdx.x & 31;
    const int wave = threadIdx.x >> 5;
    const int tile = blockIdx.x * 8 + wave;
    const bool active = tile < total_tiles;   // wave-uniform

    int b = 0, ty = 0, tx = 0;
    if (active) {
        b = tile / (tpr * tpr);
        int t2 = tile - b * tpr * tpr;
        ty = t2 / tpr;
        tx = t2 - ty * tpr;
    }
    const int y0 = ty * 16 - P;
    const int x0 = tx * 16 - P;

    if (active) {
        const float* Ib = I + (size_t)b * H * H;
        const float* Yb = (SCALE == 1) ? (Y + (size_t)b * H * H)
                                       : (Y + (size_t)b * YH * YH);
        for (int idx = lane; idx < HD * HD; idx += 32) {
            int r = idx / HD, c = idx - r * HD;
            int gh = y0 + r, gw = x0 + c;
            float vi = 0.f, vj = 0.f;
            if (gh >= 0 && gh < H && gw >= 0 && gw < H) {
                vi = Ib[gh * H + gw];
                if (SCALE == 1) {
                    vj = Yb[gh * H + gw];
                } else {
                    // half-pixel bilinear downsample: frac is exactly 0.5
                    int ly = SCALE * gh + (SCALE / 2 - 1);
                    int lx = SCALE * gw + (SCALE / 2 - 1);
                    vj = 0.25f * (Yb[ly * YH + lx]       + Yb[ly * YH + lx + 1] +
                                  Yb[(ly + 1) * YH + lx] + Yb[(ly + 1) * YH + lx + 1]);
                }
            }
            sI[wave][idx] = vi;
            sJ[wave][idx] = vj;
        }
    }
    __syncthreads();

    if (active) {
        const int half = lane >> 4;   // B/C layout: lanes 16..31 hold K+2 / M+8
        const int n    = lane & 15;   // column
        const int m    = n;           // A layout: M = lane % 16

        v8f aI = {}, aJ = {}, aI2 = {}, aJ2 = {}, aIJ = {};

        #pragma unroll
        for (int c = 0; c < CH; ++c) {
            const int r0 = 4 * c + 2 * half;  // T-row held in B vgpr0 by this lane
            const int r1 = r0 + 1;            // T-row held in B vgpr1

            float bI0 = 0.f, bJ0 = 0.f, bI20 = 0.f, bJ20 = 0.f, bIJ0 = 0.f;
            float bI1 = 0.f, bJ1 = 0.f, bI21 = 0.f, bJ21 = 0.f, bIJ1 = 0.f;
            if (r0 < HD) {
                #pragma unroll
                for (int dx = 0; dx < WIN; ++dx) {
                    float i = sI[wave][r0 * HD + n + dx];
                    float j = sJ[wave][r0 * HD + n + dx];
                    bI0 += i; bJ0 += j; bI20 += i * i; bJ20 += j * j; bIJ0 += i * j;
                }
            }
            if (r1 < HD) {
                #pragma unroll
                for (int dx = 0; dx < WIN; ++dx) {
                    float i = sI[wave][r1 * HD + n + dx];
                    float j = sJ[wave][r1 * HD + n + dx];
                    bI1 += i; bJ1 += j; bI21 += i * i; bJ21 += j * j; bIJ1 += i * j;
                }
            }
            // banded ones matrix S_v: S_v[m][k] = 1 iff m <= k <= m + 2P
            float a0 = (r0 >= m && r0 <= m + 2 * P) ? 1.f : 0.f;
            float a1 = (r1 >= m && r1 <= m + 2 * P) ? 1.f : 0.f;
            v2f A   = {a0, a1};
            v2f BI  = {bI0,  bI1},  BJ  = {bJ0,  bJ1};
            v2f BI2 = {bI20, bI21}, BJ2 = {bJ20, bJ21};
            v2f BIJ = {bIJ0, bIJ1};

            aI  = __builtin_amdgcn_wmma_f32_16x16x4_f32(false, A, false, BI,  (short)0, aI,  false, false);
            aJ  = __builtin_amdgcn_wmma_f32_16x16x4_f32(false, A, false, BJ,  (short)0, aJ,  false, false);
            aI2 = __builtin_amdgcn_wmma_f32_16x16x4_f32(false, A, false, BI2, (short)0, aI2, false, false);
            aJ2 = __builtin_amdgcn_wmma_f32_16x16x4_f32(false, A, false, BJ2, (short)0, aJ2, false, false);
            aIJ = __builtin_amdgcn_wmma_f32_16x16x4_f32(false, A, false, BIJ, (short)0, aIJ, false, false);
        }

        // Per-pixel cc + squared diff. C/D layout: vgpr r -> row r + 8*half.
        float lcc = 0.f, ll2 = 0.f;
        #pragma unroll
        for (int r = 0; r < 8; ++r) {
            float Is = aI[r], Js = aJ[r];
            float I2 = aI2[r], J2 = aJ2[r], IJ = aIJ[r];
            float uI = Is / WS, uJ = Js / WS;
            float cross = IJ - uJ * Is - uI * Js + uI * uJ * WS;
            float Iv = I2 - 2.f * uI * Is + uI * uI * WS;
            float Jv = J2 - 2.f * uJ * Js + uJ * uJ * WS;
            lcc += cross * cross / (Iv * Jv + 1e-8f);

            int mm = r + 8 * half;
            float di = sI[wave][(mm + P) * HD + (n + P)] -
                       sJ[wave][(mm + P) * HD + (n + P)];
            ll2 += di * di;
        }

        // wave32 reduction
        #pragma unroll
        for (int off = 16; off >= 1; off >>= 1) {
            lcc += __shfl_xor(lcc, off, 32);
            ll2 += __shfl_xor(ll2, off, 32);
        }
        if (lane == 0) {
            atomicAdd(accCC, lcc);
            atomicAdd(accL2, ll2);
        }
    }
}

// ---------------------------------------------------------------------------
// Combine accumulators into the 5 outputs: total, level0..level3.
//   ncc_l = -(Scc_l / B) / 100 ;  l2_l = Sl2_l / B ;  ll = w_l*(l2+ncc)/2
// ---------------------------------------------------------------------------
__global__ void hrl_finalize(const float* __restrict__ acc, float* __restrict__ out, int B) {
    if (threadIdx.x == 0) {
        const float w[4] = {1.f, 0.5f, 0.25f, 0.125f};
        float total = 0.f;
        #pragma unroll
        for (int l = 0; l < 4; ++l) {
            float Scc = acc[2 * l];
            float Sl2 = acc[2 * l + 1];
            float l2  = Sl2 / (float)B;
            float ncc = -(Scc / (float)B) / 100.f;
            float ll  = w[l] * (l2 + ncc) * 0.5f;
            out[1 + l] = ll;
            total += ll;
        }
        out[0] = total;
    }
}

extern "C" void kernel_launch(void* const* d_in, const int* in_sizes, int n_in,
                              void* d_out, int out_size, void* d_ws, size_t ws_size,
                              hipStream_t stream) {
    const float* yh0 = (const float*)d_in[0];   // [B,1,512,512]
    const float* yh1 = (const float*)d_in[1];   // [B,1,256,256]
    const float* yh2 = (const float*)d_in[2];   // [B,1,128,128]
    const float* yh3 = (const float*)d_in[3];   // [B,1,64,64]
    const float* y   = (const float*)d_in[4];   // [B,1,512,512]
    float* acc = (float*)d_ws;                  // 8 floats: (Scc, Sl2) x 4
    float* out = (float*)d_out;                 // 5 floats

    const int B = in_sizes[4] / (512 * 512);

    hrl_init_acc<<<1, 32, 0, stream>>>(acc);

    // level 0: H=512, win=9, target = y directly
    {
        int tpr = 512 / 16, total = B * tpr * tpr, blocks = (total + 7) / 8;
        hrl_level_kernel<9, 1><<<blocks, 256, 0, stream>>>(yh0, y, 512, tpr, total, acc + 0, acc + 1);
    }
    // level 1: H=256, win=9, target = 2x downsample of y
    {
        int tpr = 256 / 16, total = B * tpr * tpr, blocks = (total + 7) / 8;
        hrl_level_kernel<9, 2><<<blocks, 256, 0, stream>>>(yh1, y, 256, tpr, total, acc + 2, acc + 3);
    }
    // level 2: H=128, win=7, target = 4x downsample of y
    {
        int tpr = 128 / 16, total = B * tpr * tpr, blocks = (total + 7) / 8;
        hrl_level_kernel<7, 4><<<blocks, 256, 0, stream>>>(yh2, y, 128, tpr, total, acc + 4, acc + 5);
    }
    // level 3: H=64, win=5, target = 8x downsample of y
    {
        int tpr = 64 / 16, total = B * tpr * tpr, blocks = (total + 7) / 8;
        hrl_level_kernel<5, 8><<<blocks, 256, 0, stream>>>(yh3, y, 64, tpr, total, acc + 6, acc + 7);
    }

    hrl_finalize<<<1, 32, 0, stream>>>(acc, out, B);
}